// GRU_scratch_14250701488639
// MI455X (gfx1250) — compile-verified
//
#include <hip/hip_runtime.h>
#include <hip/hip_bf16.h>

#define HD 2048
#define TD 1024
#define BK 32
#define ASTR 36   // A LDS row stride (words): 16B-aligned rows, conflict-free frags
#define BSTR 264  // B LDS kp-row stride (words): 128 cols * 2 + 8 pad

typedef float v2f __attribute__((ext_vector_type(2)));
typedef float v8f __attribute__((ext_vector_type(8)));

// Async global->LDS staging (CDNA5): guarded so either toolchain compiles.
#if defined(__AMDGCN__) && __has_builtin(__builtin_amdgcn_global_load_async_to_lds_b128) && \
    __has_builtin(__builtin_amdgcn_global_load_async_to_lds_b32) &&                         \
    __has_builtin(__builtin_amdgcn_s_wait_asynccnt)
#define GRU_ASYNC_STAGE 1
// Param types per hipcc diagnostics:
//   b128 src: '__attribute__((__vector_size__(4*sizeof(int)))) int __device__ *'
//   b32  src: '__device__ int *'   (addrspace(1) == __device__; dst is LDS/AS3)
typedef int gi4 __attribute__((vector_size(16)));
typedef __attribute__((address_space(1))) gi4 GA_v4i;
typedef __attribute__((address_space(3))) gi4 LA_v4i;
typedef __attribute__((address_space(1))) int GA_i32;
typedef __attribute__((address_space(3))) int LA_i32;
#else
#define GRU_ASYNC_STAGE 0
#endif

// ---------------------------------------------------------------------------
// h0 = input[:, 0]
// ---------------------------------------------------------------------------
__global__ __launch_bounds__(256)
void gru_init_h(const float* __restrict__ x, float* __restrict__ h) {
  int i = blockIdx.x * blockDim.x + threadIdx.x;
  if (i < HD) h[i] = x[(size_t)i * TD];
}

// ---------------------------------------------------------------------------
// Xg = W_g @ X + b_g for g in {z,r,h}  (blockIdx.z selects gate)
// f32 WMMA 16x16x4. 64(M) x 128(N) tile per 256-thread block, K-slabs of 32,
// double-buffered LDS with async global->LDS staging (ASYNCcnt pipelined).
// B held in k-paired layout Bs[k/2][n][2] so each lane's B fragment is one
// ds_load_b64 (no register re-pairing).
// ---------------------------------------------------------------------------
__global__ __launch_bounds__(256)
void gru_wx_gemm(const float* __restrict__ Wz, const float* __restrict__ Wr,
                 const float* __restrict__ Wh, const float* __restrict__ X,
                 const float* __restrict__ bz, const float* __restrict__ br,
                 const float* __restrict__ bh, float* __restrict__ Xz,
                 float* __restrict__ Xr, float* __restrict__ Xh) {
  const int g = blockIdx.z;
  const float* __restrict__ W    = (g == 0) ? Wz : ((g == 1) ? Wr : Wh);
  const float* __restrict__ bias = (g == 0) ? bz : ((g == 1) ? br : bh);
  float* __restrict__ out        = (g == 0) ? Xz : ((g == 1) ? Xr : Xh);

  const int m0 = blockIdx.x * 64;
  const int n0 = blockIdx.y * 128;
  const int tid  = threadIdx.x;
  const int lane = tid & 31;
  const int wave = tid >> 5;
  const int wm = wave >> 1;              // 0..3 -> 16-row strip
  const int wn = wave & 1;               // 0..1 -> 64-col strip
  const int l15 = lane & 15;
  const int khalf   = (lane >> 4) << 1;  // 0 or 2 (A/B K split across half-waves)
  const int rowhalf = (lane >> 4) << 3;  // 0 or 8 (C/D M split across half-waves)

  __shared__ __align__(16) float As[2][64][ASTR];
  __shared__ __align__(16) float Bs[2][BK / 2][BSTR];

  v8f acc[4] = {};

  auto stage_slab = [&](int k0, int buf) {
#if GRU_ASYNC_STAGE
    // A slab 64x32: 8 lanes per row, 16B per lane -> b128 async (2 ops/thread)
#pragma unroll
    for (int i = 0; i < 2; ++i) {
      int idx = tid + i * 256;
      int r = idx >> 3, c4 = (idx & 7) << 2;
      __builtin_amdgcn_global_load_async_to_lds_b128(
          (GA_v4i*)(W + (size_t)(m0 + r) * HD + (k0 + c4)),
          (LA_v4i*)&As[buf][r][c4], 0, 0);
    }
    // B slab 32x128 into paired layout (element scatter) -> b32 async (16 ops)
#pragma unroll
    for (int i = 0; i < 16; ++i) {
      int idx = tid + i * 256;
      int r = idx >> 7, c = idx & 127;
      __builtin_amdgcn_global_load_async_to_lds_b32(
          (GA_i32*)(X + (size_t)(k0 + r) * TD + (n0 + c)),
          (LA_i32*)&Bs[buf][r >> 1][(c << 1) + (r & 1)], 0, 0);
    }
#else
#pragma unroll
    for (int i = 0; i < 8; ++i) {
      int idx = tid + i * 256;
      int r = idx >> 5, c = idx & 31;
      As[buf][r][c] = W[(size_t)(m0 + r) * HD + (k0 + c)];
    }
#pragma unroll
    for (int i = 0; i < 16; ++i) {
      int idx = tid + i * 256;
      int r = idx >> 7, c = idx & 127;
      Bs[buf][r >> 1][(c << 1) + (r & 1)] = X[(size_t)(k0 + r) * TD + (n0 + c)];
    }
#endif
  };

  stage_slab(0, 0);

  for (int s = 0; s < HD / BK; ++s) {
    const int buf = s & 1;
    if (s + 1 < HD / BK) {
      stage_slab((s + 1) * BK, buf ^ 1);
#if GRU_ASYNC_STAGE
      __builtin_amdgcn_s_wait_asynccnt(18);  // slab s (18 in-order ops) landed
#endif
    } else {
#if GRU_ASYNC_STAGE
      __builtin_amdgcn_s_wait_asynccnt(0);
#endif
    }
    __syncthreads();

    const v2f* __restrict__ Ap = (const v2f*)&As[buf][wm * 16 + l15][0];
#pragma unroll
    for (int k2 = 0; k2 < BK; k2 += 4) {
      const int kk = k2 + khalf;               // even
      v2f a = Ap[kk >> 1];                     // ds_load_b64
      const v2f* __restrict__ Bp = (const v2f*)&Bs[buf][kk >> 1][0];
#pragma unroll
      for (int nt = 0; nt < 4; ++nt) {
        v2f b = Bp[wn * 64 + nt * 16 + l15];   // ds_load_b64, pre-paired
        acc[nt] = __builtin_amdgcn_wmma_f32_16x16x4_f32(
            false, a, false, b, (short)0, acc[nt], false, false);
      }
    }
    __syncthreads();
  }

  // C layout: VGPR j -> M = base + rowhalf + j, N = base + l15
#pragma unroll
  for (int nt = 0; nt < 4; ++nt) {
    int col = n0 + wn * 64 + nt * 16 + l15;
#pragma unroll
    for (int j = 0; j < 8; ++j) {
      int row = m0 + wm * 16 + rowhalf + j;
      out[(size_t)row * TD + col] = acc[nt][j] + bias[row];
    }
  }
}

// ---------------------------------------------------------------------------
// wave32 reduction
// ---------------------------------------------------------------------------
__device__ __forceinline__ float wave_sum(float v) {
#pragma unroll
  for (int off = 16; off > 0; off >>= 1) v += __shfl_xor(v, off, 32);
  return v;
}

// ---------------------------------------------------------------------------
// Per-step z/r gates: az = Uz@h + Xz[:,t], ar = Ur@h + Xr[:,t]
// One wave per (gate,row) dot product; h staged in LDS once per block.
// Writes zbuf = sigmoid(az) and rhbuf = sigmoid(ar) * h.
// ---------------------------------------------------------------------------
__global__ __launch_bounds__(256)
void gru_gates_zr(const float* __restrict__ Uz, const float* __restrict__ Ur,
                  const float* __restrict__ Xz, const float* __restrict__ Xr,
                  const float* __restrict__ h, float* __restrict__ zbuf,
                  float* __restrict__ rhbuf, int t) {
  __shared__ float hs[HD];
  const int tid = threadIdx.x;
#pragma unroll
  for (int i = 0; i < 8; ++i) hs[tid + i * 256] = h[tid + i * 256];
  __syncthreads();

  const int lane = tid & 31;
  const int wave = tid >> 5;
  const int task = blockIdx.x * 8 + wave;   // 0..4095: [0,2048)=z, [2048,4096)=r
  const int gate = task >> 11;
  const int row  = task & (HD - 1);
  const float4* __restrict__ Urow =
      (const float4*)((gate ? Ur : Uz) + (size_t)row * HD);
  const float4* __restrict__ hv4 = (const float4*)hs;

  float acc = 0.f;
#pragma unroll 4
  for (int k4 = lane; k4 < HD / 4; k4 += 32) {
    __builtin_prefetch(&Urow[k4 + 64], 0, 0);   // global_prefetch_b8
    float4 u  = Urow[k4];
    float4 hv = hv4[k4];
    acc += u.x * hv.x + u.y * hv.y + u.z * hv.z + u.w * hv.w;
  }
  acc = wave_sum(acc);
  if (lane == 0) {
    const float* __restrict__ Xg = gate ? Xr : Xz;
    float a = acc + Xg[(size_t)row * TD + t];
    float s = 1.0f / (1.0f + __expf(-a));
    if (gate == 0) zbuf[row] = s;
    else           rhbuf[row] = s * hs[row];
  }
}

// ---------------------------------------------------------------------------
// Per-step h gate: hh = tanh(Uh@(r*h) + Xh[:,t]); h = z*h + (1-z)*hh
// ---------------------------------------------------------------------------
__global__ __launch_bounds__(256)
void gru_gate_h(const float* __restrict__ Uh, const float* __restrict__ Xh,
                const float* __restrict__ zbuf, const float* __restrict__ rhbuf,
                float* __restrict__ h, float* __restrict__ out, int t) {
  __shared__ float rs[HD];
  const int tid = threadIdx.x;
#pragma unroll
  for (int i = 0; i < 8; ++i) rs[tid + i * 256] = rhbuf[tid + i * 256];
  __syncthreads();

  const int lane = tid & 31;
  const int wave = tid >> 5;
  const int row  = blockIdx.x * 8 + wave;
  const float4* __restrict__ Urow = (const float4*)(Uh + (size_t)row * HD);
  const float4* __restrict__ rv4  = (const float4*)rs;

  float acc = 0.f;
#pragma unroll 4
  for (int k4 = lane; k4 < HD / 4; k4 += 32) {
    __builtin_prefetch(&Urow[k4 + 64], 0, 0);
    float4 u = Urow[k4];
    float4 r = rv4[k4];
    acc += u.x * r.x + u.y * r.y + u.z * r.z + u.w * r.w;
  }
  acc = wave_sum(acc);
  if (lane == 0) {
    float a  = acc + Xh[(size_t)row * TD + t];
    float hh = tanhf(a);
    float z  = zbuf[row];
    float hn = z * h[row] + (1.0f - z) * hh;
    h[row] = hn;
    out[(size_t)row * TD + t] = hn;
  }
}

// ---------------------------------------------------------------------------
extern "C" void kernel_launch(void* const* d_in, const int* in_sizes, int n_in,
                              void* d_out, int out_size, void* d_ws, size_t ws_size,
                              hipStream_t stream) {
  const float* input = (const float*)d_in[0];
  const float* Uz = (const float*)d_in[1];
  const float* Wz = (const float*)d_in[2];
  const float* bz = (const float*)d_in[3];
  const float* Ur = (const float*)d_in[4];
  const float* Wr = (const float*)d_in[5];
  const float* br = (const float*)d_in[6];
  const float* Uh = (const float*)d_in[7];
  const float* Wh = (const float*)d_in[8];
  const float* bh = (const float*)d_in[9];
  float* out = (float*)d_out;

  float* ws    = (float*)d_ws;
  float* Xz    = ws;
  float* Xr    = Xz + (size_t)HD * TD;
  float* Xh    = Xr + (size_t)HD * TD;
  float* hbuf  = Xh + (size_t)HD * TD;
  float* zbuf  = hbuf + HD;
  float* rhbuf = zbuf + HD;

  gru_init_h<<<HD / 256, 256, 0, stream>>>(input, hbuf);
  gru_wx_gemm<<<dim3(HD / 64, TD / 128, 3), 256, 0, stream>>>(
      Wz, Wr, Wh, input, bz, br, bh, Xz, Xr, Xh);

  for (int t = 0; t < TD; ++t) {
    gru_gates_zr<<<(2 * HD) / 8, 256, 0, stream>>>(Uz, Ur, Xz, Xr, hbuf,
                                                   zbuf, rhbuf, t);
    gru_gate_h<<<HD / 8, 256, 0, stream>>>(Uh, Xh, zbuf, rhbuf, hbuf, out, t);
  }
  (void)in_sizes; (void)n_in; (void)out_size; (void)ws_size;
}